// GCARGCN_31284541784428
// MI455X (gfx1250) — compile-verified
//
#include <hip/hip_runtime.h>

typedef __attribute__((ext_vector_type(2))) float v2f;
typedef __attribute__((ext_vector_type(8))) float v8f;

#define DD 128           // feature dim (in = hidden = out = 128)
#define PLANE_O 4112     // odd-pair plane base (float2 index): 32*128 + 16
                         //   -> +32-word bank offset vs even plane (conflict-free)

// ---------------------------------------------------------------------------
// Y[nrows,128] = act(X)[nrows,128] @ W[128,128] (+ bias[col])
//   FLAGS bit0: add bias     FLAGS bit1: apply ReLU to X on load
// One wave -> 16 rows x 128 cols via V_WMMA_F32_16X16X4_F32.
//
// fp32 WMMA VGPR layouts (CDNA5 ISA 7.12.2), half = lane/16, l16 = lane%16:
//   A 16x4 : M=l16,  K = 2*half + {0,1}   -> one float2 global load / k-step
//   B 4x16 : N=l16,  K = 2*half + {0,1}   -> one ds_load_b64 (paired W in LDS)
//   C 16x16: VGPR v -> row v + 8*half, col l16
//
// LDS holds W as float2 pairs {W[k][col], W[k+1][col]}:
//   pair p = k/2; even p at (p/2)*128+col, odd p at PLANE_O+(p/2)*128+col.
//   Half0 reads even p, half1 reads odd p; planes are 32 banks apart and
//   within a half lanes cover 32 distinct banks -> zero bank conflicts.
// ---------------------------------------------------------------------------
template<int FLAGS>
__global__ __launch_bounds__(256) void gemm128_wmma(
    const float* __restrict__ X, const float* __restrict__ W,
    const float* __restrict__ bias, float* __restrict__ Y, int nrows)
{
    __shared__ float2 lw2[PLANE_O * 2];   // 65,792 B (fits 320 KB WGP LDS)

    {   // cooperative pair-pack of W into LDS: 8192 pairs, 32 per thread
        const int t = threadIdx.x;
        #pragma unroll
        for (int i = 0; i < 32; ++i) {
            const int idx = t + i * 256;      // 0..8191
            const int p   = idx >> 7;         // pair index 0..63
            const int col = idx & 127;
            float2 v;
            v.x = W[(2 * p)     * DD + col];
            v.y = W[(2 * p + 1) * DD + col];
            const int d = (p & 1) ? (PLANE_O + (p >> 1) * DD + col)
                                  : ((p >> 1) * DD + col);
            lw2[d] = v;
        }
    }
    __syncthreads();

    const int lane = threadIdx.x & 31;
    const int tile = blockIdx.x * 8 + (threadIdx.x >> 5);  // 16-row tile
    const int m_base = tile * 16;
    if (m_base >= nrows) return;                           // wave-uniform

    const int half = lane >> 4;
    const int l16  = lane & 15;

    int arow = m_base + l16;
    if (arow >= nrows) arow = nrows - 1;                   // clamp, no branch
    const float* aptr = X + (size_t)arow * DD + 2 * half;  // K = 2*half+{0,1}
    const int bBase = half * PLANE_O + l16;                // + (k0/4)*128 + nt*16

    v8f c[8];
    #pragma unroll
    for (int i = 0; i < 8; ++i) c[i] = (v8f)0.0f;

    float2 a2 = *(const float2*)aptr;                      // prefetch k0 = 0
    #pragma unroll
    for (int k0 = 0; k0 < DD; k0 += 4) {
        float2 an = (k0 + 4 < DD) ? *(const float2*)(aptr + k0 + 4) : a2;
        if (FLAGS & 2) { a2.x = fmaxf(a2.x, 0.0f); a2.y = fmaxf(a2.y, 0.0f); }
        v2f av = { a2.x, a2.y };
        #pragma unroll
        for (int nt = 0; nt < 8; ++nt) {
            const float2 b2 = lw2[bBase + (k0 >> 2) * DD + nt * 16];
            v2f bv = { b2.x, b2.y };
            c[nt] = __builtin_amdgcn_wmma_f32_16x16x4_f32(
                false, av, false, bv, (short)0, c[nt], false, false);
        }
        a2 = an;
    }

    #pragma unroll
    for (int nt = 0; nt < 8; ++nt) {
        const int col = nt * 16 + l16;
        const float bval = (FLAGS & 1) ? bias[col] : 0.0f;
        #pragma unroll
        for (int v = 0; v < 8; ++v) {
            const int row = m_base + v + 8 * half;
            if (row < nrows)
                Y[(size_t)row * DD + col] = c[nt][v] + bval;
        }
    }
}

// ---------------------------------------------------------------------------
// For edges with etype[e]==rel:  acc[dst[e], :] += msg[src[e], :]
// One wave scans 8 edges; each lane moves a float4 slice (4 f32 atomics).
// msg (= h @ W[rel], 51.2 MB) and acc are L2-resident on the 192 MB L2.
// ---------------------------------------------------------------------------
__global__ __launch_bounds__(256) void edge_scatter(
    const float* __restrict__ msg,
    const int* __restrict__ src, const int* __restrict__ dst,
    const int* __restrict__ etype,
    float* __restrict__ acc, int nedges, int rel)
{
    const int wv   = (blockIdx.x * 256 + threadIdx.x) >> 5;
    const int lane = threadIdx.x & 31;
    const long long base = (long long)wv * 8;

    for (int j = 0; j < 8; ++j) {
        const long long e = base + j;
        if (e >= nedges) break;
        if (etype[e] != rel) continue;
        const int s = src[e];
        const int d = dst[e];
        const float4 v = *(const float4*)(msg + (size_t)s * DD + lane * 4);
        float* p = acc + (size_t)d * DD + lane * 4;
        atomicAdd(p + 0, v.x);
        atomicAdd(p + 1, v.y);
        atomicAdd(p + 2, v.z);
        atomicAdd(p + 3, v.w);
    }
}

// ---------------------------------------------------------------------------
// inputs: feats[N*D], W1[R*D*D], loop_w1[D*D], b1[D], W2[R*D*D], loop_w2[D*D],
//         b2[D], src[E], dst[E], etype[E]      output: [N*D] f32
// ws: tmp [N,128] + acc [N,128]  (~102 MB)
// ---------------------------------------------------------------------------
extern "C" void kernel_launch(void* const* d_in, const int* in_sizes, int n_in,
                              void* d_out, int out_size, void* d_ws, size_t ws_size,
                              hipStream_t stream)
{
    const float* feats  = (const float*)d_in[0];
    const float* W1     = (const float*)d_in[1];
    const float* loopw1 = (const float*)d_in[2];
    const float* b1     = (const float*)d_in[3];
    const float* W2     = (const float*)d_in[4];
    const float* loopw2 = (const float*)d_in[5];
    const float* b2     = (const float*)d_in[6];
    const int* src   = (const int*)d_in[7];
    const int* dst   = (const int*)d_in[8];
    const int* etype = (const int*)d_in[9];

    const int N = in_sizes[0] / DD;
    const int R = in_sizes[1] / (DD * DD);
    const int E = in_sizes[7];

    float* tmp = (float*)d_ws;                 // per-relation h_rel  [N,128]
    float* acc = tmp + (size_t)N * DD;         // layer-1 accumulator [N,128]
    float* out = (float*)d_out;

    const dim3 blk(256);
    const int gemmBlocks = ((N + 15) / 16 + 7) / 8;   // 8 waves / block
    const int edgeBlocks = (E + 63) / 64;             // 8 edges / wave

    // ---------------- layer 1 ----------------
    // acc = feats @ loop_w1 + b1   (self-loop + bias; full rewrite -> determinism)
    gemm128_wmma<1><<<gemmBlocks, blk, 0, stream>>>(feats, loopw1, b1, acc, N);
    for (int r = 0; r < R; ++r) {
        gemm128_wmma<0><<<gemmBlocks, blk, 0, stream>>>(
            feats, W1 + (size_t)r * DD * DD, nullptr, tmp, N);
        edge_scatter<<<edgeBlocks, blk, 0, stream>>>(
            tmp, src, dst, etype, acc, E, r);
    }

    // ---------------- layer 2 ----------------  (ReLU fused into A-loads)
    gemm128_wmma<3><<<gemmBlocks, blk, 0, stream>>>(acc, loopw2, b2, out, N);
    for (int r = 0; r < R; ++r) {
        gemm128_wmma<2><<<gemmBlocks, blk, 0, stream>>>(
            acc, W2 + (size_t)r * DD * DD, nullptr, tmp, N);
        edge_scatter<<<edgeBlocks, blk, 0, stream>>>(
            tmp, src, dst, etype, out, E, r);
    }
}